// Sinkhorn_29463475651316
// MI455X (gfx1250) — compile-verified
//
#include <hip/hip_runtime.h>
#include <math.h>

// ---------------------------------------------------------------------------
// Sinkhorn (log-domain) for MI455X / gfx1250.
//  - 16 batches of 512x512 f32, resident in d_ws (16 MB << 192 MB L2).
//  - 64 iterations x (column-LSE, row-LSE) as separate kernel launches
//    (launch order = grid-wide sync; graph-capture safe).
//  - Row-LSE uses V_WMMA_F32_16X16X4_F32 with an all-ones B operand as a
//    16-row x 4-col summation engine: D[m][*] += sum_k A[m][k]. The matrix
//    pipe does the adds while VALU does v_exp_f32 (co-execution).
// ---------------------------------------------------------------------------

typedef __attribute__((ext_vector_type(2))) float v2f;
typedef __attribute__((ext_vector_type(8))) float v8f;

#define BATCH 16
#define NF    513         // full matrix side (513)
#define D     512         // inner sub-matrix side
#define NITER 64

// ---------------------------------------------------------------------------
// Build w[b][i][j] = (i==j) ? xu[b][1+i] : 0.5*(xp[b][1+i][1+j]+xp[b][1+j][1+i])
// ---------------------------------------------------------------------------
__global__ void __launch_bounds__(256)
sinkhorn_build(const float* __restrict__ xp, const float* __restrict__ xu,
               float* __restrict__ w) {
    int idx = blockIdx.x * 256 + threadIdx.x;      // BATCH*D*D = 4,194,304
    int j = idx & (D - 1);
    int i = (idx >> 9) & (D - 1);
    int b = idx >> 18;
    const float* xpb = xp + (size_t)b * NF * NF;
    float v;
    if (i == j) {
        v = xu[b * NF + 1 + i];
    } else {
        v = 0.5f * (xpb[(size_t)(1 + i) * NF + (1 + j)] +
                    xpb[(size_t)(1 + j) * NF + (1 + i)]);
    }
    w[idx] = v;
}

// ---------------------------------------------------------------------------
// Column logsumexp (reduce over rows, axis=1):  w[:,i,j] -= lse_j
// Block: 512 threads = 128 columns x 4 row-chunks. Fully coalesced.
// Grid: 16 batches x 4 column slices = 64 blocks.
// ---------------------------------------------------------------------------
__global__ void __launch_bounds__(512)
sinkhorn_col_lse(float* __restrict__ w) {
    __shared__ float sm[4][128];
    const int b     = blockIdx.x >> 2;
    const int slice = blockIdx.x & 3;
    const int c     = threadIdx.x & 127;
    const int q     = threadIdx.x >> 7;            // row chunk 0..3
    const int j     = slice * 128 + c;
    float* __restrict__ W = w + (size_t)b * D * D + j;
    const int i0 = q * 128, i1 = i0 + 128;

    // pass 1: partial max over 128 rows
    float m = -INFINITY;
    for (int i = i0; i < i1; ++i) {
        __builtin_prefetch(W + (size_t)(i + 16) * D, 0, 0);
        m = fmaxf(m, W[(size_t)i * D]);
    }
    sm[q][c] = m;
    __syncthreads();
    m = fmaxf(fmaxf(sm[0][c], sm[1][c]), fmaxf(sm[2][c], sm[3][c]));
    __syncthreads();

    // pass 2: partial sum of exp(x - m)
    float s = 0.f;
    for (int i = i0; i < i1; ++i) s += __expf(W[(size_t)i * D] - m);
    sm[q][c] = s;
    __syncthreads();
    s = (sm[0][c] + sm[1][c]) + (sm[2][c] + sm[3][c]);
    const float lse = m + __logf(s);

    // pass 3: subtract
    for (int i = i0; i < i1; ++i) W[(size_t)i * D] -= lse;
}

// ---------------------------------------------------------------------------
// Row logsumexp (reduce over cols, axis=2):  w[:,i,:] -= lse_i
// Block: 128 threads = 4 waves = 2 panels; each panel (16 rows x 512 cols)
// is split column-wise across 2 waves. Sums via v_wmma_f32_16x16x4_f32 with
// B = ones; A tile per ISA layout: lanes 0-15 hold K=0,1; lanes 16-31 K=2,3.
// Grid: 512 panels / 2 = 256 blocks.
// ---------------------------------------------------------------------------
__global__ void __launch_bounds__(128)
sinkhorn_row_lse(float* __restrict__ w) {
    __shared__ float sMax[2][2][16];   // [panelInBlock][waveHalf][row]
    __shared__ float sSum[2][2][16];
    __shared__ float sLse[2][16];

    const int wv   = threadIdx.x >> 5;   // 0..3
    const int lane = threadIdx.x & 31;
    const int p    = wv >> 1;            // panel within block
    const int h    = wv & 1;             // column half (0: cols 0..255, 1: 256..511)
    const int half = lane >> 4;          // A-tile K split
    const int r    = lane & 15;          // row within panel (M = lane%16)

    const int gpanel = blockIdx.x * 2 + p;      // 0..511
    const int b      = gpanel >> 5;
    const int panel  = gpanel & 31;
    float* __restrict__ Wp   = w + ((size_t)b * D + (size_t)panel * 16) * D;
    float* __restrict__ Wrow = Wp + (size_t)r * D;
    const int cbase = h * 256;

    // Phase A: per-row max over this wave's 256 columns (2 lanes per row)
    float m = -INFINITY;
    for (int c = cbase + half * 4; c < cbase + 256; c += 8) {
        float4 v = *(const float4*)(Wrow + c);
        m = fmaxf(m, fmaxf(fmaxf(v.x, v.y), fmaxf(v.z, v.w)));
    }
    m = fmaxf(m, __shfl_xor(m, 16, 32));
    if (half == 0) sMax[p][h][r] = m;
    __syncthreads();
    m = fmaxf(sMax[p][0][r], sMax[p][1][r]);    // full row max (uniform per row)

    // Phase B: WMMA row-sums of exp(x - m).  A tile: lane<16 -> (K=0,1),
    // lane>=16 -> (K=2,3) at M = lane%16.  B = all-ones (layout-agnostic).
    v8f acc = {0.f, 0.f, 0.f, 0.f, 0.f, 0.f, 0.f, 0.f};
    v2f ones; ones[0] = 1.0f; ones[1] = 1.0f;
    const int co = half * 2;
    for (int c0 = cbase; c0 < cbase + 256; c0 += 4) {
        v2f a;
        a[0] = __expf(Wrow[c0 + co]     - m);
        a[1] = __expf(Wrow[c0 + co + 1] - m);
        acc = __builtin_amdgcn_wmma_f32_16x16x4_f32(
                  false, a, false, ones, (short)0, acc, false, false);
    }

    // Extract row sums from the C/D layout: VGPR g, lanes 0-15 -> M=g,
    // lanes 16-31 -> M=g+8 (all N columns identical since B was ones).
    if (lane == 0) {
#pragma unroll
        for (int g = 0; g < 8; ++g) sSum[p][h][g] = acc[g];
    } else if (lane == 16) {
#pragma unroll
        for (int g = 0; g < 8; ++g) sSum[p][h][8 + g] = acc[g];
    }
    __syncthreads();
    if (h == 0 && half == 0) {
        sLse[p][r] = m + __logf(sSum[p][0][r] + sSum[p][1][r]);
    }
    __syncthreads();

    // Phase C: coalesced subtract over this wave's column half
    for (int rr = 0; rr < 16; ++rr) {
        const float l = sLse[p][rr];
        float* __restrict__ Wr = Wp + (size_t)rr * D;
        for (int c = cbase + lane; c < cbase + 256; c += 32) Wr[c] -= l;
    }
}

// ---------------------------------------------------------------------------
// Finalize xp: corner -> exp, first row/col -> passthrough, interior ->
// exp(w) off-diagonal, 0 on diagonal.
// ---------------------------------------------------------------------------
__global__ void __launch_bounds__(256)
sinkhorn_final_xp(const float* __restrict__ xp_in, const float* __restrict__ w,
                  float* __restrict__ out) {
    int idx = blockIdx.x * 256 + threadIdx.x;
    if (idx >= BATCH * NF * NF) return;
    int j = idx % NF;
    int t = idx / NF;
    int i = t % NF;
    int b = t / NF;
    float v;
    if (i == 0 && j == 0) {
        v = __expf(xp_in[idx]);
    } else if (i == 0 || j == 0) {
        v = xp_in[idx];
    } else if (i == j) {
        v = 0.0f;
    } else {
        v = __expf(w[((size_t)b * D + (i - 1)) * D + (j - 1)]);
    }
    out[idx] = v;
}

// ---------------------------------------------------------------------------
// Finalize xu: [0] -> exp(input), [1:] -> exp(diag(w))
// ---------------------------------------------------------------------------
__global__ void __launch_bounds__(256)
sinkhorn_final_xu(const float* __restrict__ xu_in, const float* __restrict__ w,
                  float* __restrict__ out) {
    int idx = blockIdx.x * 256 + threadIdx.x;
    if (idx >= BATCH * NF) return;
    int k = idx % NF;
    int b = idx / NF;
    float v;
    if (k == 0) {
        v = __expf(xu_in[idx]);
    } else {
        v = __expf(w[((size_t)b * D + (k - 1)) * D + (k - 1)]);
    }
    out[idx] = v;
}

// ---------------------------------------------------------------------------
extern "C" void kernel_launch(void* const* d_in, const int* in_sizes, int n_in,
                              void* d_out, int out_size, void* d_ws, size_t ws_size,
                              hipStream_t stream) {
    (void)in_sizes; (void)n_in; (void)out_size; (void)ws_size;
    const float* xp_in = (const float*)d_in[0];   // (16, 513, 513) f32
    const float* xu_in = (const float*)d_in[1];   // (16, 513)      f32
    float* w   = (float*)d_ws;                    // 16 MB workspace, L2-resident
    float* out = (float*)d_out;                   // xp (16*513*513) ++ xu (16*513)

    sinkhorn_build<<<(BATCH * D * D) / 256, 256, 0, stream>>>(xp_in, xu_in, w);

    for (int it = 0; it < NITER; ++it) {
        sinkhorn_col_lse<<<BATCH * 4, 512, 0, stream>>>(w);   // axis=1 first
        sinkhorn_row_lse<<<(BATCH * 32) / 2, 128, 0, stream>>>(w); // then axis=2
    }

    sinkhorn_final_xp<<<(BATCH * NF * NF + 255) / 256, 256, 0, stream>>>(xp_in, w, out);
    sinkhorn_final_xu<<<(BATCH * NF + 255) / 256, 256, 0, stream>>>(
        xu_in, w, out + (size_t)BATCH * NF * NF);
}